// Constraint1_61701500175357
// MI455X (gfx1250) — compile-verified
//
#include <hip/hip_runtime.h>
#include <math.h>

// Problem constants from the reference: D is [262144, 512] float32.
#define N_ROWS      262144
#define N_COLS      512
#define STRIPS      (N_COLS / 16)        // 32 column strips of 16
#define WPS         256                  // waves (row segments) per strip
#define ROWS_PER_WAVE (N_ROWS / WPS)     // 1024 rows per wave
#define WAVES_PER_BLOCK 8                // 256 threads, wave32

typedef __attribute__((ext_vector_type(2))) float v2f;
typedef __attribute__((ext_vector_type(8))) float v8f;

// Gram-trick column-norm accumulator.
// Each wave: strip of 16 columns, ROWS_PER_WAVE rows, 4 rows per WMMA.
// Lane L (L<16)  loads rows r+0, r+1 of column strip*16 + L
// Lane L (L>=16) loads rows r+2, r+3 of column strip*16 + (L-16)
// v_wmma_f32_16x16x4_f32 with SRC0 == SRC1 => C diagonal accumulates
// sum of squares of the 4 values held by lanes j and j+16 (column j of strip).
__global__ __launch_bounds__(256) void colnorm_wmma_kernel(
    const float* __restrict__ D, float* __restrict__ part) {
  const int lane  = threadIdx.x & 31;
  const int wave  = threadIdx.x >> 5;
  const int strip = blockIdx.x;                       // 0..31
  const int wslot = blockIdx.y * WAVES_PER_BLOCK + wave;  // 0..WPS-1
  const int r0    = wslot * ROWS_PER_WAVE;

  const int col      = lane & 15;
  const int row_half = (lane >> 4) * 2;               // 0 or 2

  const float* p = D + (size_t)(r0 + row_half) * N_COLS + strip * 16 + col;

  v8f c = {};  // 16x16 f32 accumulator (diagonal is what we want)

  // 16 rows (4 WMMAs) per iteration; ROWS_PER_WAVE/16 iterations.
  for (int it = 0; it < ROWS_PER_WAVE / 16; ++it) {
    v2f a0, a1, a2, a3;
    // Streaming data, touched once: non-temporal to spare the 192MB L2.
    a0[0] = __builtin_nontemporal_load(p + 0 * 4 * N_COLS);
    a0[1] = __builtin_nontemporal_load(p + 0 * 4 * N_COLS + N_COLS);
    a1[0] = __builtin_nontemporal_load(p + 1 * 4 * N_COLS);
    a1[1] = __builtin_nontemporal_load(p + 1 * 4 * N_COLS + N_COLS);
    a2[0] = __builtin_nontemporal_load(p + 2 * 4 * N_COLS);
    a2[1] = __builtin_nontemporal_load(p + 2 * 4 * N_COLS + N_COLS);
    a3[0] = __builtin_nontemporal_load(p + 3 * 4 * N_COLS);
    a3[1] = __builtin_nontemporal_load(p + 3 * 4 * N_COLS + N_COLS);

    // (neg_a, A, neg_b, B, c_mod, C, reuse_a, reuse_b)
    c = __builtin_amdgcn_wmma_f32_16x16x4_f32(false, a0, false, a0, (short)0, c, false, false);
    c = __builtin_amdgcn_wmma_f32_16x16x4_f32(false, a1, false, a1, (short)0, c, false, false);
    c = __builtin_amdgcn_wmma_f32_16x16x4_f32(false, a2, false, a2, (short)0, c, false, false);
    c = __builtin_amdgcn_wmma_f32_16x16x4_f32(false, a3, false, a3, (short)0, c, false, false);

    p += 16 * N_COLS;
  }

  // Extract diagonal of C.
  // C layout: VGPR g, lanes 0-15 -> (M=g, N=lane); lanes 16-31 -> (M=8+g, N=lane-16).
  // Diag j (j<8)  lives at lane j,     VGPR j.
  // Diag j (j>=8) lives at lane 16+j = 24+(j-8), VGPR j-8.
  float diagv = 0.0f;
#pragma unroll
  for (int g = 0; g < 8; ++g) {
    if (lane == g)      diagv = c[g];   // columns 0..7
    if (lane == 24 + g) diagv = c[g];   // columns 8..15
  }

  if (lane < 8 || lane >= 24) {
    const int c_idx = (lane < 8) ? lane : (lane - 16);
    part[((size_t)strip * WPS + wslot) * 16 + c_idx] = diagv;
  }
}

// Deterministic finish: fold WPS partials per column, residual vs 1,
// Frobenius norm of the 512 residuals, scale by 0.001.
__global__ __launch_bounds__(512) void finish_kernel(
    const float* __restrict__ part, float* __restrict__ out) {
  __shared__ float s[512];
  const int j = threadIdx.x;          // column 0..511
  const int strip = j >> 4;
  const int col   = j & 15;

  const float* p = part + (size_t)strip * WPS * 16 + col;
  float sum = 0.0f;
  for (int w = 0; w < WPS; ++w) sum += p[(size_t)w * 16];

  const float r = sum - 1.0f;
  s[j] = r * r;
  __syncthreads();

  for (int off = 256; off > 0; off >>= 1) {
    if (j < off) s[j] += s[j + off];
    __syncthreads();
  }
  if (j == 0) out[0] = 0.001f * sqrtf(s[0]);
}

extern "C" void kernel_launch(void* const* d_in, const int* in_sizes, int n_in,
                              void* d_out, int out_size, void* d_ws, size_t ws_size,
                              hipStream_t stream) {
  (void)in_sizes; (void)n_in; (void)out_size; (void)ws_size;
  const float* D   = (const float*)d_in[0];
  float*       out = (float*)d_out;
  float*       part = (float*)d_ws;   // STRIPS*WPS*16 floats = 512 KB, fully overwritten

  dim3 grid(STRIPS, WPS / WAVES_PER_BLOCK);   // (32, 32) -> 8192 waves
  dim3 block(32 * WAVES_PER_BLOCK);           // 256 threads = 8 wave32s
  colnorm_wmma_kernel<<<grid, block, 0, stream>>>(D, part);
  finish_kernel<<<1, 512, 0, stream>>>(part, out);
}